// TotalLoss_7868380086555
// MI455X (gfx1250) — compile-verified
//
#include <hip/hip_runtime.h>
#include <stdint.h>

// ---------------- problem constants (fixed by reference shapes) -------------
#define EPS_F      1.1920929e-07f   // float32 machine eps
#define CHAR_EPS2  1.0e-06f         // CHAR_EPS^2
#define CHAR_ALPHA 0.45f
#define SMOOTH_W   0.5f

#define COEFF_BASE 16               // float offset of coeff table in ws
#define ACC_BASE   512              // float offset of accumulators in ws
// per-scale accumulator bases (floats), B=8, 8*HW per (b): 8*8*HW per scale
// HW = 1024,4096,16384,65536 ; bases: 0, 65536, 327680, 1376256 ; total 5570560

typedef float v2f __attribute__((ext_vector_type(2)));
typedef float v8f __attribute__((ext_vector_type(8)));

// Low 32 bits of a flat LDS pointer are the wave-relative LDS byte offset
// (ISA 10.2 aperture mapping), which is exactly what VDST of the async op wants.
__device__ __forceinline__ unsigned lds_off(const void* p) {
    return (unsigned)(uintptr_t)p;
}
__device__ __forceinline__ void async_load_f32(unsigned lds_dst, const float* src) {
    unsigned long long ga = (unsigned long long)(uintptr_t)src;
    asm volatile("global_load_async_to_lds_b32 %0, %1, off"
                 :: "v"(lds_dst), "v"(ga) : "memory");
}
__device__ __forceinline__ void wait_async0() {
    asm volatile("s_wait_asynccnt 0" ::: "memory");
}

// ---------------- 0) zero workspace ----------------------------------------
__global__ void zero_k(float* __restrict__ ws, size_t n) {
    size_t stride = (size_t)gridDim.x * blockDim.x;
    for (size_t i = (size_t)blockIdx.x * blockDim.x + threadIdx.x; i < n; i += stride)
        ws[i] = 0.f;
}

// ---------------- 1) per-batch time-transform coefficients -----------------
// t is renormalized each scale: t_s = A*t + B with
//   A' = A/dn, B' = (B - t0)/dn, dn = tN - t0 + EPS  (t0/tN from OLD transform)
// Then per (pol,dir): t_ = (tEnd - B +/- EPS) + (-A)*t
__global__ void ev_prep(const float* __restrict__ events, float* __restrict__ ws, int N) {
    int b = blockIdx.x;
    const float* t = events + ((size_t)b * 4 + 2) * N;
    const float* p = events + ((size_t)b * 4 + 3) * N;
    __shared__ int smn[2], smx[2];
    if (threadIdx.x < 2) { smn[threadIdx.x] = N; smx[threadIdx.x] = -1; }
    __syncthreads();
    int mn0 = N, mx0 = -1, mn1 = N, mx1 = -1;
    for (int i = threadIdx.x; i < N; i += blockDim.x) {
        float pv = p[i];
        if (pv > 0.f) { mn0 = min(mn0, i); mx0 = max(mx0, i); }
        else          { mn1 = min(mn1, i); mx1 = max(mx1, i); }
    }
    atomicMin(&smn[0], mn0); atomicMax(&smx[0], mx0);
    atomicMin(&smn[1], mn1); atomicMax(&smx[1], mx1);
    __syncthreads();
    if (threadIdx.x == 0) {
        int fp = smn[0] == N ? 0 : smn[0];
        int lp = smx[0] <  0 ? N - 1 : smx[0];
        int fn = smn[1] == N ? 0 : smn[1];
        int ln = smx[1] <  0 ? N - 1 : smx[1];
        float tf = t[0], tl = t[N - 1];
        float tfp = t[fp], tlp = t[lp], tfn = t[fn], tln = t[ln];
        float A = 1.f, Bc = 0.f;
        float* cbase = ws + COEFF_BASE + b * 32;
        for (int s = 0; s < 4; ++s) {
            float t0 = A * tf + Bc, tN = A * tl + Bc;
            float dn = tN - t0 + EPS_F;
            A = A / dn; Bc = (Bc - t0) / dn;
            float t0p = A * tfp + Bc, tNp = A * tlp + Bc;
            float t0n = A * tfn + Bc, tNn = A * tln + Bc;
            float* c = cbase + s * 8;
            c[0] = tNp - Bc + EPS_F;  c[1] = -A;   // pos, forward:  tN - t + EPS
            c[2] = t0p - Bc - EPS_F;  c[3] = -A;   // pos, backward: t0 - t - EPS
            c[4] = tNn - Bc + EPS_F;  c[5] = -A;   // neg, forward
            c[6] = t0n - Bc - EPS_F;  c[7] = -A;   // neg, backward
        }
    }
}

// ---------------- 2) event scatter (async-LDS staged, L2-resident atomics) -
__global__ void ev_scatter(const float* __restrict__ events,
                           const float* __restrict__ f0, const float* __restrict__ f1,
                           const float* __restrict__ f2, const float* __restrict__ f3,
                           float* __restrict__ ws, int N) {
    __shared__ float sx[256], sy[256], st[256], sp[256];
    const int tid = threadIdx.x;
    const int s   = blockIdx.y & 3;
    const int b   = blockIdx.y >> 2;
    const int w   = 32 << s;
    const int hw  = w * w;
    const float invScale = (float)(1 << s) * 0.125f;   // 1 / 2^(3-s)
    const float* flow = (s == 0) ? f0 : (s == 1) ? f1 : (s == 2) ? f2 : f3;
    const float* fxp = flow + (size_t)b * 2 * hw;
    const float* fyp = fxp + hw;
    const size_t scaleBase = (s == 0) ? 0u : (s == 1) ? 65536u : (s == 2) ? 327680u : 1376256u;
    float* accb = ws + ACC_BASE + scaleBase + (size_t)b * 8 * hw;
    const float* cc = ws + COEFF_BASE + b * 32 + s * 8;
    float c00 = cc[0], c01 = cc[1], c02 = cc[2], c03 = cc[3];
    float c10 = cc[4], c11 = cc[5], c12 = cc[6], c13 = cc[7];
    const float* ex = events + (size_t)b * 4 * N;
    const float* ey = ex + N;
    const float* et = ex + 2 * (size_t)N;
    const float* ep = ex + 3 * (size_t)N;
    unsigned lx = lds_off(&sx[tid]), ly = lds_off(&sy[tid]);
    unsigned lt = lds_off(&st[tid]), lp = lds_off(&sp[tid]);
    const float wm1 = (float)(w - 1);

    for (int tile = blockIdx.x; tile * 256 < N; tile += gridDim.x) {
        int i  = tile * 256 + tid;
        int ic = i < N ? i : N - 1;
        async_load_f32(lx, ex + ic);
        async_load_f32(ly, ey + ic);
        async_load_f32(lt, et + ic);
        async_load_f32(lp, ep + ic);
        wait_async0();                       // own-lane LDS slots now valid
        float x = sx[tid], y = sy[tid], t = st[tid], p = sp[tid];
        if (i < N) {
            int pol = p > 0.f ? 0 : 1;
            int xi = (int)(x * invScale);
            int yi = (int)(y * invScale);
            int fo = yi * w + xi;
            float fx = fxp[fo], fy = fyp[fo];
            float d0c0 = pol ? c10 : c00, d0c1 = pol ? c11 : c01;
            float d1c0 = pol ? c12 : c02, d1c1 = pol ? c13 : c03;
            float* base = accb + (size_t)pol * 4 * hw;   // (pol*2+dir)*2*hw blocks
            #pragma unroll
            for (int dir = 0; dir < 2; ++dir) {
                float t_ = dir ? (d1c0 + d1c1 * t) : (d0c0 + d0c1 * t);
                float x_ = fminf(fmaxf((float)xi + t_ * fx, 0.f), wm1);
                float y_ = fminf(fmaxf((float)yi + t_ * fy, 0.f), wm1);
                float x0 = floorf(x_), x1 = ceilf(x_);
                float y0 = floorf(y_), y1 = ceilf(y_);
                float x0r = 1.f - (x_ - x0), x1r = 1.f - (x1 - x_);
                float y0r = 1.f - (y_ - y0), y1r = 1.f - (y1 - y_);
                int ix0 = (int)x0, ix1 = (int)x1, iy0 = (int)y0, iy1 = (int)y1;
                float* num = base + (size_t)dir * 2 * hw;
                float* den = num + hw;
                int i00 = iy0 * w + ix0, i10 = iy0 * w + ix1;
                int i01 = iy1 * w + ix0, i11 = iy1 * w + ix1;
                atomicAdd(den + i00, 1.f); atomicAdd(num + i00, (x0r * y0r + EPS_F) * t_);
                atomicAdd(den + i10, 1.f); atomicAdd(num + i10, (x1r * y0r + EPS_F) * t_);
                atomicAdd(den + i01, 1.f); atomicAdd(num + i01, (x0r * y1r + EPS_F) * t_);
                atomicAdd(den + i11, 1.f); atomicAdd(num + i11, (x1r * y1r + EPS_F) * t_);
            }
        }
    }
}

// ---------------- 3) sum (num/(den+eps))^2 with WMMA wave reduction --------
// A(16x4) = lane partials (one nonzero/lane), B(4x16) = ones -> every column of
// D holds the 16 row sums; lanes 0-15 carry rows 0-7, lanes 16-31 rows 8-15.
__global__ void ev_reduce(float* __restrict__ ws) {
    const size_t TL = 2785280u;   // 32 combos * (1024+4096+16384+65536)
    size_t stride = (size_t)gridDim.x * blockDim.x;
    float part = 0.f;
    for (size_t idx = (size_t)blockIdx.x * blockDim.x + threadIdx.x; idx < TL; idx += stride) {
        int s; size_t r;
        if      (idx < 32768u)  { s = 0; r = idx; }
        else if (idx < 163840u) { s = 1; r = idx - 32768u; }
        else if (idx < 688128u) { s = 2; r = idx - 163840u; }
        else                    { s = 3; r = idx - 688128u; }
        const int shift = 10 + 2 * s;
        const size_t hw = (size_t)1 << shift;
        int combo = (int)(r >> shift);
        size_t j = r & (hw - 1);
        int b = combo >> 2, pd = combo & 3;
        size_t scaleBase = (s == 0) ? 0u : (s == 1) ? 65536u : (s == 2) ? 327680u : 1376256u;
        const float* base = ws + ACC_BASE + scaleBase + ((size_t)b * 8 + (size_t)pd * 2) * hw;
        float nu = base[j], de = base[hw + j];
        float v = nu / (de + EPS_F);
        part += v * v;
    }
    v2f a;  a.x = part; a.y = 0.f;
    v2f bm; bm.x = 1.f; bm.y = 1.f;
    v8f c = {0.f, 0.f, 0.f, 0.f, 0.f, 0.f, 0.f, 0.f};
    c = __builtin_amdgcn_wmma_f32_16x16x4_f32(false, a, false, bm, (short)0, c, false, false);
    float wsum = c[0] + c[1] + c[2] + c[3] + c[4] + c[5] + c[6] + c[7];
    wsum += __shfl_xor(wsum, 16);                 // rows 0-7 + rows 8-15
    if ((threadIdx.x & 31) == 0) atomicAdd(ws + 0, wsum);
}

// ---------------- 4) Charbonnier smoothness --------------------------------
__global__ void smooth_k(const float* __restrict__ flow, int planes, int h, int w,
                         float* __restrict__ ws) {
    size_t total  = (size_t)planes * h * w;
    size_t stride = (size_t)gridDim.x * blockDim.x;
    float inv0 = 1.f / ((float)planes * h * (w - 1));        // along-w diffs
    float inv1 = 1.f / ((float)planes * (h - 1) * w);        // along-h diffs
    float inv2 = 1.f / ((float)planes * (h - 1) * (w - 1));  // both diagonals
    float acc = 0.f;
    for (size_t idx = (size_t)blockIdx.x * blockDim.x + threadIdx.x; idx < total; idx += stride) {
        int j = (int)(idx % (size_t)w);
        int i = (int)((idx / (size_t)w) % (size_t)h);
        const float* f = flow + idx;
        float v = f[0];
        if (j + 1 < w) { float d = f[1] - v;     acc += powf(d * d + CHAR_EPS2, CHAR_ALPHA) * inv0; }
        if (i + 1 < h) { float d = f[w] - v;     acc += powf(d * d + CHAR_EPS2, CHAR_ALPHA) * inv1; }
        if (i + 1 < h && j + 1 < w) {
            float d  = f[w + 1] - v;             acc += powf(d  * d  + CHAR_EPS2, CHAR_ALPHA) * inv2;
            float d2 = f[1] - f[w];              acc += powf(d2 * d2 + CHAR_EPS2, CHAR_ALPHA) * inv2;
        }
    }
    for (int off = 16; off > 0; off >>= 1) acc += __shfl_xor(acc, off);
    if ((threadIdx.x & 31) == 0) atomicAdd(ws + 1, acc * SMOOTH_W);
}

// ---------------- 5) finalize ----------------------------------------------
__global__ void fin_k(const float* __restrict__ ws, float* __restrict__ out) {
    if (threadIdx.x == 0) {
        float ev = ws[0], sm = ws[1];
        out[0] = ev + sm;   // loss
        out[1] = ev;        // event_loss
        out[2] = sm;        // smooth
    }
}

extern "C" void kernel_launch(void* const* d_in, const int* in_sizes, int n_in,
                              void* d_out, int out_size, void* d_ws, size_t ws_size,
                              hipStream_t stream) {
    (void)n_in; (void)out_size; (void)ws_size;
    const float* events = (const float*)d_in[0];
    const float* f0 = (const float*)d_in[1];
    const float* f1 = (const float*)d_in[2];
    const float* f2 = (const float*)d_in[3];
    const float* f3 = (const float*)d_in[4];
    float* ws  = (float*)d_ws;
    float* out = (float*)d_out;
    const int N = in_sizes[0] / (8 * 4);   // events: (B,4,N)

    zero_k<<<2048, 256, 0, stream>>>(ws, (size_t)(ACC_BASE + 5570560));
    ev_prep<<<8, 256, 0, stream>>>(events, ws, N);
    dim3 g((N + 255) / 256, 32);           // y = batch*4 + scale combos (8*4)
    ev_scatter<<<g, 256, 0, stream>>>(events, f0, f1, f2, f3, ws, N);
    ev_reduce<<<1024, 256, 0, stream>>>(ws);
    smooth_k<<<256,  256, 0, stream>>>(f0, 16, 32,  32,  ws);
    smooth_k<<<256,  256, 0, stream>>>(f1, 16, 64,  64,  ws);
    smooth_k<<<512,  256, 0, stream>>>(f2, 16, 128, 128, ws);
    smooth_k<<<1024, 256, 0, stream>>>(f3, 16, 256, 256, ws);
    fin_k<<<1, 1, 0, stream>>>(ws, out);
}